// GraphConvMonteCarlo_74208444940819
// MI455X (gfx1250) — compile-verified
//
#include <hip/hip_runtime.h>
#include <hip/hip_bf16.h>

// ---------------------------------------------------------------------------
// GraphConvMonteCarlo fused kernel for MI455X (gfx1250, wave32, WMMA).
//
//   out[N,128] = A[N,896] @ W[896,128]
//   A row n = concat over edge_type et of:
//     et<6 : mean of 2 threefry-sampled neighbor rows x[col[g]] (0 if empty)
//     et==6: x[n]
//
// Kernel 1 (optional, needs 448KB of d_ws): swizzle W into K-panels of 4 so a
// WMMA B fragment is a single global_load_b64 from a contiguous 256B region.
// Kernel 2: per block, stage a 16x896 fp32 A-tile in LDS, then 8 waves each
// drive 224x V_WMMA_F32_16X16X4_F32 for a 16x16 C tile.
// ---------------------------------------------------------------------------

typedef float v2f __attribute__((ext_vector_type(2)));
typedef float v8f __attribute__((ext_vector_type(8)));

#define C_IN   128
#define C_OUT  128
#define NET    7
#define KDIM   (NET * C_IN)      // 896
#define LDA    (KDIM + 4)        // 900-float A row pitch (bank stagger, 16B aligned)
#define TILE_M 16
#define NROWS  (TILE_M * NET)    // 112 slot-rows per tile

// ------------------------- threefry-2x32 (JAX PRNG) ------------------------

__host__ __device__ __forceinline__ unsigned rotl32(unsigned x, int r) {
  return (x << r) | (x >> (32 - r));
}

__host__ __device__ __forceinline__ void threefry2x32(unsigned k0, unsigned k1,
                                                      unsigned c0, unsigned c1,
                                                      unsigned* o0, unsigned* o1) {
  unsigned ks2 = 0x1BD11BDAu ^ k0 ^ k1;
  unsigned x0 = c0 + k0, x1 = c1 + k1;
#define TF_G(a, b, c, d)                                                       \
  x0 += x1; x1 = rotl32(x1, a); x1 ^= x0;                                      \
  x0 += x1; x1 = rotl32(x1, b); x1 ^= x0;                                      \
  x0 += x1; x1 = rotl32(x1, c); x1 ^= x0;                                      \
  x0 += x1; x1 = rotl32(x1, d); x1 ^= x0;
  TF_G(13, 15, 26, 6)  x0 += k1;  x1 += ks2 + 1u;
  TF_G(17, 29, 16, 24) x0 += ks2; x1 += k0  + 2u;
  TF_G(13, 15, 26, 6)  x0 += k0;  x1 += k1  + 3u;
  TF_G(17, 29, 16, 24) x0 += k1;  x1 += ks2 + 4u;
  TF_G(13, 15, 26, 6)  x0 += ks2; x1 += k0  + 5u;
#undef TF_G
  *o0 = x0; *o1 = x1;
}

// jax random_bits(key, 32, (S,2)) flat element f; counters [0,2S) split in
// halves: pair i = (i, i+S) -> (bits[i], bits[i+S]).
__device__ __forceinline__ unsigned jax_bits(unsigned key0, unsigned key1,
                                             int f, int S) {
  unsigned o0, o1;
  if (f < S) { threefry2x32(key0, key1, (unsigned)f, (unsigned)(f + S), &o0, &o1); return o0; }
  threefry2x32(key0, key1, (unsigned)(f - S), (unsigned)f, &o0, &o1);
  return o1;
}

__device__ __forceinline__ int lower_bound(const int* __restrict__ a, int n, int v) {
  int lo = 0, hi = n;
  while (lo < hi) {
    int mid = (lo + hi) >> 1;
    if (a[mid] < v) lo = mid + 1; else hi = mid;
  }
  return lo;
}

// ----------------- kernel 1: W -> K-panel-of-4 swizzle (d_ws) ---------------
// Wswz[(k>>2)*512 + n*4 + (k&3)] = W[k*128 + n]
// Phase-2 B fragment then reads 2 consecutive floats per lane; a full wave's
// fragment is one contiguous, fully-used 256B region per WMMA.

__global__ __launch_bounds__(256) void
swizzle_W_kernel(const float* __restrict__ W, float* __restrict__ Wswz) {
  const int t = blockIdx.x * 256 + threadIdx.x;   // over KDIM*C_OUT
  if (t >= KDIM * C_OUT) return;
  const int n = t & (C_OUT - 1);
  const int k = t >> 7;
  Wswz[(k >> 2) * (C_OUT * 4) + n * 4 + (k & 3)] = W[t];
}

// ------------------------------- main kernel -------------------------------

__global__ __launch_bounds__(256) void
graphconv_mc_wmma(const float* __restrict__ x, const int* __restrict__ col,
                  const int* __restrict__ idx_sorted,
                  const float* __restrict__ W,       // original [896][128]
                  const float* __restrict__ Wswz,    // swizzled (may be null)
                  float* __restrict__ out, int N, int E, int useSwz,
                  unsigned key0, unsigned key1) {
  __shared__ float Atile[TILE_M * LDA];   // 57.6 KB fp32 A tile
  __shared__ int metaMode[NROWS];         // 0 = empty slot, 1 = self, 2 = pair
  __shared__ int metaC0[NROWS];
  __shared__ int metaC1[NROWS];

  const int node0 = blockIdx.x * TILE_M;
  const int tid   = threadIdx.x;
  const int wave  = tid >> 5;
  const int lane  = tid & 31;
  const int S     = N * NET;

  // ---- Phase 1a: slot metadata (segment search + threefry sampling) ----
  if (tid < NROWS) {
    const int m = tid / NET, et = tid % NET;
    const int node = node0 + m;
    int mode = 0, c0 = 0, c1 = 0;
    if (node < N) {
      if (et == NET - 1) {               // self slot
        mode = 1; c0 = node;
      } else {
        const int s  = node * NET + et;
        const int lo = lower_bound(idx_sorted, E, s);
        const int hi = lower_bound(idx_sorted, E, s + 1);
        const int cnt = hi - lo;
        if (cnt > 0) {
          // randint(0,2^30): span 2^30 reduces JAX's hi/lo recombination to
          // lower_bits & 0x3FFFFFFF, key = split(key(42))[1].
          const unsigned b0 = jax_bits(key0, key1, 2 * s,     S);
          const unsigned b1 = jax_bits(key0, key1, 2 * s + 1, S);
          const int r0 = (int)(b0 & 0x3FFFFFFFu) % cnt;
          const int r1 = (int)(b1 & 0x3FFFFFFFu) % cnt;
          c0 = col[lo + r0];
          c1 = col[lo + r1];
          mode = 2;
        }
      }
    }
    metaMode[tid] = mode; metaC0[tid] = c0; metaC1[tid] = c1;
  }
  __syncthreads();

  // ---- Phase 1b: gather/average x rows into the A tile (float4/lane) ----
  for (int r = wave; r < NROWS; r += 8) {
    const int mode = metaMode[r];
    const int m = r / NET, et = r % NET;
    float4 v = make_float4(0.f, 0.f, 0.f, 0.f);
    if (mode == 1) {
      v = ((const float4*)(x + (size_t)metaC0[r] * C_IN))[lane];
    } else if (mode == 2) {
      const float4 a = ((const float4*)(x + (size_t)metaC0[r] * C_IN))[lane];
      const float4 b = ((const float4*)(x + (size_t)metaC1[r] * C_IN))[lane];
      v = make_float4(0.5f * (a.x + b.x), 0.5f * (a.y + b.y),
                      0.5f * (a.z + b.z), 0.5f * (a.w + b.w));
    }
    ((float4*)(Atile + m * LDA + et * C_IN))[lane] = v;
  }
  __syncthreads();

  // ---- Phase 2: 16x16 C tile per wave, K=896 via V_WMMA_F32_16X16X4_F32 ----
  // A frag (ISA 16x4 f32): lanes 0-15 row M=lane hold K{0,1}; lanes 16-31 row
  // M=lane-16 hold K{2,3} -> ds_load_b64 per lane. B frag mirrors on K.
  const int nSub    = lane & 15;
  const int kHalf   = lane >> 4;
  const int colBase = wave * 16;
  const float* aBase = Atile + nSub * LDA + kHalf * 2;

  v8f acc = {0.f, 0.f, 0.f, 0.f, 0.f, 0.f, 0.f, 0.f};

  if (useSwz) {
    // Swizzled: lane's B frag = 2 consecutive floats; stride 2048B per k-step,
    // all loop addressing folds into immediate offsets.
    const float* bBase = Wswz + (colBase + nSub) * 4 + kHalf * 2;
#pragma unroll 8
    for (int k0 = 0; k0 < KDIM; k0 += 4) {
      const v2f a = *(const v2f*)(aBase + k0);
      const v2f b = *(const v2f*)(bBase + k0 * C_OUT);   // (k0/4)*512 floats
      acc = __builtin_amdgcn_wmma_f32_16x16x4_f32(false, a, false, b,
                                                  (short)0, acc, false, false);
    }
  } else {
    const float* wBase = W + colBase + nSub;
#pragma unroll 8
    for (int k0 = 0; k0 < KDIM; k0 += 4) {
      const v2f a = *(const v2f*)(aBase + k0);
      v2f b;
      b.x = wBase[(k0 + kHalf * 2 + 0) * C_OUT];
      b.y = wBase[(k0 + kHalf * 2 + 1) * C_OUT];
      acc = __builtin_amdgcn_wmma_f32_16x16x4_f32(false, a, false, b,
                                                  (short)0, acc, false, false);
    }
  }

  // ---- Epilogue: D VGPR j -> (M = j + 8*kHalf, N = nSub); 8 consecutive
  // rows per lane -> single base + immediate offsets on the full-tile path.
  float* oBase = out + (size_t)(node0 + kHalf * 8) * C_OUT + colBase + nSub;
  if (node0 + TILE_M <= N) {
#pragma unroll
    for (int j = 0; j < 8; ++j) oBase[(size_t)j * C_OUT] = acc[j];
  } else {
#pragma unroll
    for (int j = 0; j < 8; ++j)
      if (node0 + kHalf * 8 + j < N) oBase[(size_t)j * C_OUT] = acc[j];
  }
}

// ------------------------------ host launcher ------------------------------

extern "C" void kernel_launch(void* const* d_in, const int* in_sizes, int n_in,
                              void* d_out, int out_size, void* d_ws, size_t ws_size,
                              hipStream_t stream) {
  const float* x          = (const float*)d_in[0];
  const int*   col        = (const int*)d_in[1];
  const int*   idx_sorted = (const int*)d_in[2];
  const float* W          = (const float*)d_in[3];
  float*       out        = (float*)d_out;

  const int N = in_sizes[0] / C_IN;   // 100000
  const int E = in_sizes[1];          // 700000

  // jax.random.split(key(42))[1]: threefry((0,42), iota(4)) ->
  // bits = [o0_p0, o0_p1, o1_p0, o1_p1]; second subkey = (o1_p0, o1_p1).
  unsigned o0a, o1a, o0b, o1b;
  threefry2x32(0u, 42u, 0u, 2u, &o0a, &o1a);
  threefry2x32(0u, 42u, 1u, 3u, &o0b, &o1b);
  const unsigned key0 = o1a, key1 = o1b;

  const size_t wNeed = (size_t)KDIM * C_OUT * sizeof(float);   // 448 KB
  const int useSwz = (d_ws != nullptr && ws_size >= wNeed) ? 1 : 0;
  float* Wswz = (float*)d_ws;

  if (useSwz) {
    const int totW = KDIM * C_OUT;
    hipLaunchKernelGGL(swizzle_W_kernel, dim3((totW + 255) / 256), dim3(256),
                       0, stream, W, Wswz);
  }

  const int blocks = (N + TILE_M - 1) / TILE_M;   // 6250
  hipLaunchKernelGGL(graphconv_mc_wmma, dim3(blocks), dim3(256), 0, stream,
                     x, col, idx_sorted, W, Wswz, out, N, E, useSwz, key0, key1);
}